// EnhancedGCNN_69286412419205
// MI455X (gfx1250) — compile-verified
//
#include <hip/hip_runtime.h>
#include <cstdint>
#include <cstddef>

typedef __attribute__((ext_vector_type(2))) float v2f;
typedef __attribute__((ext_vector_type(8))) float v8f;

#define D_H    128
#define BN_EPS 1e-5f

// ---------------------------------------------------------------------------
// C[M x 128] = A[M x 128] * W[128 x 128]  (+bias per column, optional ReLU)
// One wave32 computes one 16x16 tile via 32x V_WMMA_F32_16X16X4_F32.
// Block = 128 threads = 4 waves; gridDim.x = M/16, gridDim.y = 2 (8 n-tiles).
// Requires M % 16 == 0 (true: 100000 and 2048).
// ---------------------------------------------------------------------------
__global__ __launch_bounds__(128) void gemm128_wmma(
    const float* __restrict__ A, const float* __restrict__ W,
    float* __restrict__ C, const float* __restrict__ bias, int relu) {
  const int wave  = threadIdx.x >> 5;
  const int lane  = threadIdx.x & 31;
  const int half  = lane >> 4;     // 0: lanes 0-15, 1: lanes 16-31
  const int l     = lane & 15;
  const int mbase = blockIdx.x * 16;
  const int nbase = (blockIdx.y * 4 + wave) * 16;

  const float* arow = A + (size_t)(mbase + l) * D_H;
  v8f acc = {};
#pragma unroll
  for (int k = 0; k < D_H; k += 4) {
    const int ka = k + 2 * half;          // this lane-half supplies K=ka, ka+1
    v2f a, b;
    a[0] = arow[ka];
    a[1] = arow[ka + 1];
    b[0] = W[(size_t)ka * D_H + nbase + l];
    b[1] = W[(size_t)(ka + 1) * D_H + nbase + l];
    acc = __builtin_amdgcn_wmma_f32_16x16x4_f32(
        /*neg_a=*/false, a, /*neg_b=*/false, b,
        /*c_mod=*/(short)0, acc, /*reuse_a=*/false, /*reuse_b=*/false);
  }
  const float bv = bias ? bias[nbase + l] : 0.0f;
#pragma unroll
  for (int j = 0; j < 8; ++j) {
    float v = acc[j] + bv;                // D VGPR j -> row M = j + 8*half
    if (relu) v = fmaxf(v, 0.0f);
    C[(size_t)(mbase + j + 8 * half) * D_H + nbase + l] = v;
  }
}

// ---------------------------------------------------------------------------
// Degree / normalization
// ---------------------------------------------------------------------------
__global__ void deg_init(float* __restrict__ deg, int N) {
  int i = blockIdx.x * blockDim.x + threadIdx.x;
  if (i < N) deg[i] = 1.0f;               // self-loop contributes 1
}
__global__ void deg_edges(const int* __restrict__ dst, float* __restrict__ deg, int E) {
  int e = blockIdx.x * blockDim.x + threadIdx.x;
  if (e < E) atomicAdd(&deg[dst[e]], 1.0f);
}
__global__ void deg_rsqrt(float* __restrict__ deg, int N) {
  int i = blockIdx.x * blockDim.x + threadIdx.x;
  if (i < N) deg[i] = rsqrtf(deg[i]);     // deg >= 1 always (self-loops)
}

// ---------------------------------------------------------------------------
// out[i] = dinv[i]^2 * h[i] + bias   (self-loop message + conv bias)
// Threads cover N*32 float4 groups.
// ---------------------------------------------------------------------------
__global__ void selfloop_bias_init(const float* __restrict__ h,
                                   const float* __restrict__ dinv,
                                   const float* __restrict__ bias,
                                   float* __restrict__ out, int N) {
  int idx = blockIdx.x * blockDim.x + threadIdx.x;
  int total = N * (D_H / 4);
  if (idx >= total) return;
  int i  = idx >> 5;                      // node
  int f4 = idx & 31;                      // float4 group within row
  float di = dinv[i];
  float s  = di * di;
  float4 hv = ((const float4*)(h + (size_t)i * D_H))[f4];
  float4 bv = ((const float4*)bias)[f4];
  float4 o;
  o.x = hv.x * s + bv.x;  o.y = hv.y * s + bv.y;
  o.z = hv.z * s + bv.z;  o.w = hv.w * s + bv.w;
  ((float4*)(out + (size_t)i * D_H))[f4] = o;
}

// ---------------------------------------------------------------------------
// Edge scatter: out[dst] += dinv[src]*dinv[dst]*h[src]. One wave per edge,
// lane does one float4 gather (coalesced 512B/wave) + 4 float atomics (L2).
// ---------------------------------------------------------------------------
__global__ __launch_bounds__(256) void aggregate_edges(
    const float* __restrict__ h, const int* __restrict__ src,
    const int* __restrict__ dst, const float* __restrict__ dinv,
    float* __restrict__ out, int E) {
  int gw   = (blockIdx.x * blockDim.x + threadIdx.x) >> 5;
  int lane = threadIdx.x & 31;
  if (gw >= E) return;
  int s = src[gw];
  int d = dst[gw];
  float norm = dinv[s] * dinv[d];
  float4 v = ((const float4*)(h + (size_t)s * D_H))[lane];
  float* op = out + (size_t)d * D_H + lane * 4;
  atomicAdd(op + 0, v.x * norm);
  atomicAdd(op + 1, v.y * norm);
  atomicAdd(op + 2, v.z * norm);
  atomicAdd(op + 3, v.w * norm);
}

// ---------------------------------------------------------------------------
// In-place BN(eval) + ReLU over N x 128
// ---------------------------------------------------------------------------
__global__ void bn_relu(float* __restrict__ h, const float* __restrict__ g,
                        const float* __restrict__ b, const float* __restrict__ m,
                        const float* __restrict__ v, int N) {
  int idx = blockIdx.x * blockDim.x + threadIdx.x;
  if (idx >= N * D_H) return;
  int f = idx & (D_H - 1);
  float s = g[f] * rsqrtf(v[f] + BN_EPS);
  h[idx] = fmaxf((h[idx] - m[f]) * s + b[f], 0.0f);
}

// ---------------------------------------------------------------------------
// Mean pool over graphs
// ---------------------------------------------------------------------------
__global__ void pool_zero(float* __restrict__ sums, float* __restrict__ cnts, int G) {
  int idx = blockIdx.x * blockDim.x + threadIdx.x;
  if (idx < G * D_H) sums[idx] = 0.0f;
  if (idx < G) cnts[idx] = 0.0f;
}
__global__ __launch_bounds__(256) void pool_nodes(
    const float* __restrict__ h, const int* __restrict__ batch,
    float* __restrict__ sums, float* __restrict__ cnts, int N) {
  int gw   = (blockIdx.x * blockDim.x + threadIdx.x) >> 5;
  int lane = threadIdx.x & 31;
  if (gw >= N) return;
  int g = batch[gw];
  float4 v = ((const float4*)(h + (size_t)gw * D_H))[lane];
  float* sp = sums + (size_t)g * D_H + lane * 4;
  atomicAdd(sp + 0, v.x);
  atomicAdd(sp + 1, v.y);
  atomicAdd(sp + 2, v.z);
  atomicAdd(sp + 3, v.w);
  if (lane == 0) atomicAdd(&cnts[g], 1.0f);
}
__global__ void pool_div(float* __restrict__ sums, const float* __restrict__ cnts, int G) {
  int idx = blockIdx.x * blockDim.x + threadIdx.x;
  if (idx >= G * D_H) return;
  sums[idx] /= fmaxf(cnts[idx / D_H], 1.0f);
}

// ---------------------------------------------------------------------------
// logits = z @ fc2_W + fc2_b  (D_OUT = 2), then log_softmax. 1 thread/graph.
// ---------------------------------------------------------------------------
__global__ void fc2_logsoftmax(const float* __restrict__ z,
                               const float* __restrict__ W,
                               const float* __restrict__ b,
                               float* __restrict__ out, int G) {
  int g = blockIdx.x * blockDim.x + threadIdx.x;
  if (g >= G) return;
  float l0 = b[0], l1 = b[1];
  const float* zp = z + (size_t)g * D_H;
#pragma unroll 8
  for (int k = 0; k < D_H; ++k) {
    float zv = zp[k];
    l0 += zv * W[2 * k + 0];
    l1 += zv * W[2 * k + 1];
  }
  float mx  = fmaxf(l0, l1);
  float lse = mx + logf(expf(l0 - mx) + expf(l1 - mx));
  out[2 * g + 0] = l0 - lse;
  out[2 * g + 1] = l1 - lse;
}

// ---------------------------------------------------------------------------
extern "C" void kernel_launch(void* const* d_in, const int* in_sizes, int n_in,
                              void* d_out, int out_size, void* d_ws, size_t ws_size,
                              hipStream_t stream) {
  (void)n_in; (void)ws_size;

  const float* x     = (const float*)d_in[0];
  const int*   eidx  = (const int*)d_in[1];
  const int*   batch = (const int*)d_in[2];
  const float* Ws[3] = {(const float*)d_in[3], (const float*)d_in[5], (const float*)d_in[7]};
  const float* bs[3] = {(const float*)d_in[4], (const float*)d_in[6], (const float*)d_in[8]};
  const float *bng[3], *bnb[3], *bnm[3], *bnv[3];
  for (int i = 0; i < 3; ++i) {
    bng[i] = (const float*)d_in[9 + 4 * i + 0];
    bnb[i] = (const float*)d_in[9 + 4 * i + 1];
    bnm[i] = (const float*)d_in[9 + 4 * i + 2];
    bnv[i] = (const float*)d_in[9 + 4 * i + 3];
  }
  const float* fc1_W = (const float*)d_in[21];
  const float* fc1_b = (const float*)d_in[22];
  const float* fc2_W = (const float*)d_in[23];
  const float* fc2_b = (const float*)d_in[24];

  const int N = in_sizes[0] / D_H;   // 100000
  const int E = in_sizes[1] / 2;     // 3200000
  const int G = out_size / 2;        // 2048
  const int* src = eidx;
  const int* dst = eidx + E;

  // --- workspace carve-out ---
  char* ws = (char*)d_ws;
  size_t off = 0;
  auto carve = [&](size_t bytes) {
    void* p = ws + off;
    off += (bytes + 255) & ~(size_t)255;
    return p;
  };
  float* dinv = (float*)carve((size_t)N * sizeof(float));
  float* hA   = (float*)carve((size_t)N * D_H * sizeof(float));
  float* hB   = (float*)carve((size_t)N * D_H * sizeof(float));
  float* sums = (float*)carve((size_t)G * D_H * sizeof(float));
  float* cnts = (float*)carve((size_t)G * sizeof(float));
  float* z    = (float*)carve((size_t)G * D_H * sizeof(float));

  // --- degrees -> dinv ---
  deg_init<<<(N + 255) / 256, 256, 0, stream>>>(dinv, N);
  deg_edges<<<(E + 255) / 256, 256, 0, stream>>>(dst, dinv, E);
  deg_rsqrt<<<(N + 255) / 256, 256, 0, stream>>>(dinv, N);

  // --- 3 GCN layers (ping-pong: in -> hA (gemm) -> hB (aggregate+bn)) ---
  const float* in = x;
  for (int L = 0; L < 3; ++L) {
    gemm128_wmma<<<dim3(N / 16, 2), 128, 0, stream>>>(in, Ws[L], hA, nullptr, 0);
    selfloop_bias_init<<<(N * 32 + 255) / 256, 256, 0, stream>>>(hA, dinv, bs[L], hB, N);
    aggregate_edges<<<(E * 32 + 255) / 256, 256, 0, stream>>>(hA, src, dst, dinv, hB, E);
    bn_relu<<<(N * D_H + 255) / 256, 256, 0, stream>>>(hB, bng[L], bnb[L], bnm[L], bnv[L], N);
    in = hB;
  }

  // --- global mean pool ---
  pool_zero<<<(G * D_H + 255) / 256, 256, 0, stream>>>(sums, cnts, G);
  pool_nodes<<<(N * 32 + 255) / 256, 256, 0, stream>>>(hB, batch, sums, cnts, N);
  pool_div<<<(G * D_H + 255) / 256, 256, 0, stream>>>(sums, cnts, G);

  // --- FC1 (WMMA, fused bias+ReLU) ---
  gemm128_wmma<<<dim3(G / 16, 2), 128, 0, stream>>>(sums, fc1_W, z, fc1_b, 1);

  // --- FC2 + log_softmax ---
  fc2_logsoftmax<<<(G + 255) / 256, 256, 0, stream>>>(z, fc2_W, fc2_b, (float*)d_out, G);
}